// tensor_DLT_5016521802530
// MI455X (gfx1250) — compile-verified
//
#include <hip/hip_runtime.h>

#define BLOCK 256

// Closed-form 4-point DLT homography (Heckbert square->quad + adjugate inverse).
// One batch per thread; LDS-staged coalesced output via CDNA5 async LDS->global stores.
__global__ __launch_bounds__(BLOCK)
void dlt_homography_kernel(const float* __restrict__ in, float* __restrict__ out, int nb) {
    __shared__ float sout[BLOCK * 9];
    const int tid = threadIdx.x;
    const int idx = blockIdx.x * BLOCK + tid;

    // ---- load 8 floats per batch as two b128 loads (coalesced) ----
    float4 lo = make_float4(0.f, 0.f, 0.f, 0.f);
    float4 hi = lo;
    if (idx < nb) {
        const float4* in4 = (const float4*)in;
        lo = in4[2 * idx + 0];
        hi = in4[2 * idx + 1];
    }

    const float P = 128.0f;
    // orig corners (x_k, y_k); canonical corners are (0,0),(128,0),(0,128),(128,128)
    float x0 = lo.x,     y0 = lo.y;
    float x1 = lo.z + P, y1 = lo.w;
    float x2 = hi.x,     y2 = hi.y + P;
    float x3 = hi.z + P, y3 = hi.w + P;

    // ---- Heckbert: unit square (0,0),(1,0),(0,1),(1,1) -> quad p0..p3 ----
    float dx1 = x1 - x3, dx2 = x2 - x3;
    float dy1 = y1 - y3, dy2 = y2 - y3;
    float sx  = (x0 - x1) - (x2 - x3);
    float sy  = (y0 - y1) - (y2 - y3);
    float det  = dx1 * dy2 - dx2 * dy1;
    float rdet = 1.0f / det;
    float g = (sx * dy2 - sy * dx2) * rdet;
    float h = (dx1 * sy - dy1 * sx) * rdet;

    float A  = fmaf(x1, g, x1 - x0);
    float Bq = fmaf(x2, h, x2 - x0);
    float D  = fmaf(y1, g, y1 - y0);
    float E  = fmaf(y2, h, y2 - y0);

    // Compose with diag(1/128, 1/128, 1): square [0,128]^2 -> quad
    const float s = 0.0078125f; // 1/128
    float m00 = A * s,  m01 = Bq * s, m02 = x0;
    float m10 = D * s,  m11 = E * s,  m12 = y0;
    float m20 = g * s,  m21 = h * s;  // m22 = 1

    // ---- adjugate (m22 == 1); H = cof^T / cof22 so H[2][2] == 1 ----
    float c00 = fmaf(-m12, m21, m11);
    float c01 = fmaf( m12, m20, -m10);
    float c02 = m10 * m21 - m11 * m20;
    float c10 = fmaf( m02, m21, -m01);
    float c11 = fmaf(-m02, m20, m00);
    float c12 = -(m00 * m21 - m01 * m20);
    float c20 = m01 * m12 - m02 * m11;
    float c21 = -(m00 * m12 - m02 * m10);
    float c22 = m00 * m11 - m01 * m10;

    float rn = 1.0f / c22;

    float* w = &sout[tid * 9];   // stride 9 (odd) -> bank-conflict-free
    w[0] = c00 * rn;  w[1] = c10 * rn;  w[2] = c20 * rn;
    w[3] = c01 * rn;  w[4] = c11 * rn;  w[5] = c21 * rn;
    w[6] = c02 * rn;  w[7] = c12 * rn;  w[8] = 1.0f;

    __syncthreads();

    // ---- coalesced writeout: 2304 floats = 576 float4 per block ----
    const long long base  = (long long)blockIdx.x * (BLOCK * 9);
    const long long limit = (long long)nb * 9 - base;
    if (limit >= (long long)(BLOCK * 9)) {
        float* gbase = out + base;
        for (int j = tid; j < (BLOCK * 9) / 4; j += BLOCK) {
            void*    gaddr   = (void*)(gbase + 4 * j);
            unsigned lds_off = (unsigned)(unsigned long long)(void*)(sout + 4 * j);
            // CDNA5 async LDS->global DMA store, tracked by ASYNCcnt
            asm volatile("global_store_async_from_lds_b128 %0, %1, off"
                         :
                         : "v"(gaddr), "v"(lds_off)
                         : "memory");
        }
        asm volatile("s_wait_asynccnt 0" ::: "memory");
    } else {
        // tail block (not hit for B = 524288): plain guarded stores
        for (long long j = tid; j < limit; j += BLOCK)
            out[base + j] = sout[j];
    }
}

extern "C" void kernel_launch(void* const* d_in, const int* in_sizes, int n_in,
                              void* d_out, int out_size, void* d_ws, size_t ws_size,
                              hipStream_t stream) {
    const float* in  = (const float*)d_in[0];
    float*       out = (float*)d_out;
    const int nb = in_sizes[0] / 8;           // B batches of 8 floats
    const int grid = (nb + BLOCK - 1) / BLOCK;
    dlt_homography_kernel<<<grid, BLOCK, 0, stream>>>(in, out, nb);
}